// YoloLoss_25838523253136
// MI455X (gfx1250) — compile-verified
//
#include <hip/hip_runtime.h>
#include <math.h>

// Problem constants (must match reference)
#define SS 13
#define BBX 5
#define NC 20
#define MM (SS * SS * BBX)   // 845 boxes per sample
#define REC (5 + NC)         // 25 floats per box record
#define NTHREADS 512
#define NWAVES (NTHREADS / 32)  // 16 wave32s

typedef float v2f __attribute__((ext_vector_type(2)));
typedef float v8f __attribute__((ext_vector_type(8)));

__global__ __launch_bounds__(NTHREADS)
void yolo_v2_loss_kernel(const float* __restrict__ yhat,
                         const float* __restrict__ ytrue,
                         const float* __restrict__ anchors,
                         const int*   __restrict__ epoch_p,
                         float*       __restrict__ out,
                         int N, int out_size)
{
  __shared__ float4 hc[MM];        // pred corners  (x0,y0,x1,y1)
  __shared__ float4 tc[MM];        // truth corners
  __shared__ float  ha[MM];        // pred areas
  __shared__ float  ta[MM];        // truth areas
  __shared__ float  anc[BBX * 2];  // anchors
  __shared__ float  wsum[NWAVES][16]; // per-wave partials, padded to 16 cols

  const int n    = blockIdx.x;
  const int t    = threadIdx.x;
  const int lane = t & 31;
  const int wave = t >> 5;

  const float* Yh = yhat  + (size_t)n * MM * REC;
  const float* Yt = ytrue + (size_t)n * MM * REC;

  // Pull the whole sample toward the caches (gfx1250 global_prefetch_b8);
  // pass B gathers 25-float records at data-dependent indices.
  for (int off = t * 64; off < MM * REC; off += NTHREADS * 64) {
    __builtin_prefetch(Yh + off, 0, 0);
    __builtin_prefetch(Yt + off, 0, 0);
  }

  if (t < BBX * 2) anc[t] = anchors[t];

  // ---- Stage corners + areas into LDS ----
  for (int i = t; i < MM; i += NTHREADS) {
    int   cell = i / BBX;
    float gx = (float)(cell % SS);
    float gy = (float)(cell / SS);
    {
      const float* p = Yh + i * REC;
      float cx = (p[0] + gx) / (float)SS;
      float cy = (p[1] + gy) / (float)SS;
      float w = p[2], h = p[3];
      float x0 = cx - 0.5f * w, y0 = cy - 0.5f * h;
      float x1 = cx + 0.5f * w, y1 = cy + 0.5f * h;
      hc[i] = make_float4(x0, y0, x1, y1);
      ha[i] = (x1 - x0) * (y1 - y0);
    }
    {
      const float* p = Yt + i * REC;
      float cx = (p[0] + gx) / (float)SS;
      float cy = (p[1] + gy) / (float)SS;
      float w = p[2], h = p[3];
      float x0 = cx - 0.5f * w, y0 = cy - 0.5f * h;
      float x1 = cx + 0.5f * w, y1 = cy + 0.5f * h;
      tc[i] = make_float4(x0, y0, x1, y1);
      ta[i] = (x1 - x0) * (y1 - y0);
    }
  }
  __syncthreads();

  // ---- Pass A: per pred box i, does any true box give IoU >= 0.5 ? ----
  // iou >= 0.5  <=>  2*inter >= union + eps   (no division needed)
  float noobj_p = 0.0f;
  for (int i = t; i < MM; i += NTHREADS) {
    float4 a  = hc[i];
    float aar = ha[i];
    int hit = 0;
    for (int j = 0; j < MM; ++j) {
      float4 b = tc[j];
      float wi = fminf(a.z, b.z) - fmaxf(a.x, b.x);
      float hi = fminf(a.w, b.w) - fmaxf(a.y, b.y);
      wi = fmaxf(wi, 0.0f);
      hi = fmaxf(hi, 0.0f);
      float inter = wi * hi;
      float uni   = aar + ta[j] - inter;
      hit |= (2.0f * inter >= uni + 1e-16f) ? 1 : 0;
    }
    if (!hit) {  // no_obj_iou mask (max_iou < IOU_THRES), LAMBDA_NOOBJ = 1
      float d = Yh[i * REC + 4] - Yt[i * REC + 4];
      noobj_p += d * d;
    }
  }

  // ---- Pass B: per true slot j, argmax over pred boxes i (first-max wins) ----
  // iou_i > iou_best  <=>  inter_i*(union_best+eps) > inter_best*(union_i+eps)
  float coord_p = 0.0f, class_p = 0.0f, obj_p = 0.0f, prior_p = 0.0f;
  const int do_prior = (epoch_p[0] < 20) ? 1 : 0;  // EPOCH_PRIOR

  for (int j = t; j < MM; j += NTHREADS) {
    float4 b  = tc[j];
    float bar = ta[j];
    int   bidx   = 0;
    float binter = -1.0f, buni = 0.0f;  // first compare always accepts i=0
    for (int i = 0; i < MM; ++i) {
      float4 a = hc[i];
      float wi = fminf(a.z, b.z) - fmaxf(a.x, b.x);
      float hi = fminf(a.w, b.w) - fmaxf(a.y, b.y);
      wi = fmaxf(wi, 0.0f);
      hi = fmaxf(hi, 0.0f);
      float inter = wi * hi;
      float uni   = ha[i] + bar - inter;
      if (inter * (buni + 1e-16f) > binter * (uni + 1e-16f)) {
        binter = inter; buni = uni; bidx = i;
      }
    }

    const float* pt = Yt + j * REC;
    const float* ph = Yh + bidx * REC;   // gathered yhat_res
    float y2 = pt[2], y3 = pt[3], y4 = pt[4];
    float have = (y4 > 0.0f) ? 1.0f : 0.0f;
    int ab = (bidx % BBX) * 2;   // anchor of the gathered pred box
    int jb = (j    % BBX) * 2;   // anchor of this slot

    float dx = ph[0] - pt[0];
    float dy = ph[1] - pt[1];
    float dw = logf(ph[2] / anc[ab + 0] + 1e-16f) - logf(y2 / anc[jb + 0] + 1e-16f);
    float dh = logf(ph[3] / anc[ab + 1] + 1e-16f) - logf(y3 / anc[jb + 1] + 1e-16f);
    coord_p += (dx * dx + dy * dy + dw * dw + dh * dh) * have * (2.0f - y2 * y3);

    float cls = 0.0f;
    for (int c = 0; c < NC; ++c) {
      float dc = ph[5 + c] - pt[5 + c];
      cls += dc * dc;
    }
    class_p += cls * have;            // LAMBDA_CLASS = 1

    float dob = ph[4] - y4;
    obj_p += dob * dob * have;        // LAMBDA_OBJ applied below

    if (do_prior) {
      const float* pj = Yh + j * REC;
      float pw  = pj[2] - anc[jb + 0];
      float phh = pj[3] - anc[jb + 1];
      prior_p += pw * pw + phh * phh; // LAMBDA_PRIOR applied below
    }
  }
  obj_p   *= 5.0f;
  prior_p *= 0.01f;

  // ---- Intra-wave reduction (wave32 xor-shuffles) ----
  float v0 = coord_p, v1 = class_p, v2 = noobj_p, v3 = obj_p, v4 = prior_p;
  #pragma unroll
  for (int off = 16; off >= 1; off >>= 1) {
    v0 += __shfl_xor(v0, off, 32);
    v1 += __shfl_xor(v1, off, 32);
    v2 += __shfl_xor(v2, off, 32);
    v3 += __shfl_xor(v3, off, 32);
    v4 += __shfl_xor(v4, off, 32);
  }
  if (lane == 0) {
    wsum[wave][0] = v0; wsum[wave][1] = v1; wsum[wave][2] = v2;
    wsum[wave][3] = v3; wsum[wave][4] = v4;
    #pragma unroll
    for (int q = 5; q < 16; ++q) wsum[wave][q] = 0.0f;
  }
  __syncthreads();

  // ---- Cross-wave reduction as a real ones-GEMM on the matrix core ----
  // A = ones(16x4), B[k][col] = wave-partials, C-chained:
  // D[m][col] = sum over all 16 waves of wsum[w][col]. Column index is
  // lane%16 in every documented CDNA5 layout, and a column-sum is invariant
  // to the K-row permutation across lane halves / VGPRs.
  if (t < 32) {   // wave 0, EXEC all ones as WMMA requires
    v2f av; av[0] = 1.0f; av[1] = 1.0f;   // A all ones
    v8f acc = {};
    int col  = lane & 15;
    int half = lane >> 4;
    #pragma unroll
    for (int g = 0; g < NWAVES; g += 4) {
      v2f bv;
      bv[0] = wsum[g + half * 2 + 0][col];
      bv[1] = wsum[g + half * 2 + 1][col];
      acc = __builtin_amdgcn_wmma_f32_16x16x4_f32(
          /*neg_a=*/false, av, /*neg_b=*/false, bv,
          /*c_mod=*/(short)0, acc, /*reuse_a=*/false, /*reuse_b=*/false);
    }
    float r = acc[0];   // D[M=0][N=col] lives in VGPR0 of lanes 0..15
    if (lane < 4) {
      out[lane * N + n] = r;          // coord, class, no_obj, obj
    } else if (lane == 4) {
      int o = 4 * N + n;              // prior (shape (N,) or (1,))
      if (o < out_size) out[o] = r;
    }
  }
}

extern "C" void kernel_launch(void* const* d_in, const int* in_sizes, int n_in,
                              void* d_out, int out_size, void* d_ws, size_t ws_size,
                              hipStream_t stream) {
  const float* yhat    = (const float*)d_in[0];
  const float* y       = (const float*)d_in[1];
  const float* anchors = (const float*)d_in[2];
  const int*   epoch   = (const int*)d_in[3];
  int N = in_sizes[0] / (SS * SS * BBX * REC);   // 32 for the given setup
  yolo_v2_loss_kernel<<<dim3(N), dim3(NTHREADS), 0, stream>>>(
      yhat, y, anchors, epoch, (float*)d_out, N, out_size);
}